// CorrespondenceContrastiveLoss_44873818309267
// MI455X (gfx1250) — compile-verified
//
#include <hip/hip_runtime.h>

// Feature volume geometry: [1, C=64, X=100, Y=88, Z=80]
#define CH_STRIDE 704000   // 100*88*80, stride between channels (elements)
#define SX 7040            // 88*80
#define SY 80

typedef float v2f __attribute__((ext_vector_type(2)));
typedef float v8f __attribute__((ext_vector_type(8)));

__device__ __forceinline__ int pymod(int a, int m) {
    int r = a % m;
    return (r < 0) ? r + m : r;   // Python-style mod (divisor sign)
}

// One wave handles 16 points. Lane L (L<16) accumulates channels [0,32) of
// point L; lane L+16 accumulates channels [32,64) of point L. The two halves
// are summed by a single V_WMMA_F32_16X16X4_F32 against an all-ones B matrix:
// A-layout (ISA 7.12.2, 32-bit A 16x4): row M = {v0,v1}@lane M  U  {v0,v1}@lane M+16,
// so with v1=0 the row-sum D[M][*] = partial(M) + partial(M+16) = full distance.
__global__ __launch_bounds__(256) void ccl_main(
    const float* __restrict__ fixf,
    const float* __restrict__ movf,
    const int*   __restrict__ fpts,
    const int*   __restrict__ ppts,
    const int*   __restrict__ npts,
    float*       __restrict__ partials)
{
    const int tid  = threadIdx.x;
    const int lane = tid & 31;
    const int wave = tid >> 5;
    const int half = lane >> 4;                              // channel half
    const int pidx = (blockIdx.x * 8 + wave) * 16 + (lane & 15);

    // Redirect points: mod(pt - CROP_INDEX, CROP_HALF), Python semantics.
    const int fx = pymod(fpts[pidx * 3 + 0] - 25,  100);
    const int fy = pymod(fpts[pidx * 3 + 1] - 225, 88);
    const int fz = pymod(fpts[pidx * 3 + 2] - 28,  80);
    const int px = pymod(ppts[pidx * 3 + 0] - 25,  100);
    const int py = pymod(ppts[pidx * 3 + 1] - 225, 88);
    const int pz = pymod(ppts[pidx * 3 + 2] - 28,  80);
    const int nx = pymod(npts[pidx * 3 + 0] - 25,  100);
    const int ny = pymod(npts[pidx * 3 + 1] - 225, 88);
    const int nz = pymod(npts[pidx * 3 + 2] - 28,  80);

    const long cofs = (long)half * 32 * CH_STRIDE;
    const float* __restrict__ fb = fixf + (fx * SX + fy * SY + fz) + cofs;
    const float* __restrict__ pb = movf + (px * SX + py * SY + pz) + cofs;
    const float* __restrict__ nb = movf + (nx * SX + ny * SY + nz) + cofs;

    // 32 channels per lane, 3 gathers each; unroll for memory-level parallelism
    // (gather-latency bound: keep many independent global_load_b32 in flight).
    float accp = 0.0f, accn = 0.0f;
    #pragma unroll 8
    for (int j = 0; j < 32; ++j) {
        const long o = (long)j * CH_STRIDE;
        const float a = fb[o];
        const float b = pb[o];
        const float g = nb[o];
        const float t1 = a - b;
        const float t2 = a - g;
        accp = fmaf(t1, t1, accp);
        accn = fmaf(t2, t2, accn);
    }

    // Cross-lane channel reduction via WMMA row-sum (B = ones, C = 0).
    v2f a_p; a_p[0] = accp; a_p[1] = 0.0f;
    v2f a_n; a_n[0] = accn; a_n[1] = 0.0f;
    v2f ones; ones[0] = 1.0f; ones[1] = 1.0f;
    v8f zero = {};
    v8f dpos = __builtin_amdgcn_wmma_f32_16x16x4_f32(
        false, a_p, false, ones, (short)0, zero, false, false);
    v8f dneg = __builtin_amdgcn_wmma_f32_16x16x4_f32(
        false, a_n, false, ones, (short)0, zero, false, false);

    // D layout: VGPR r holds point r (lanes 0-15) / point r+8 (lanes 16-31),
    // replicated across columns. Each half-wave reduces its 8 points (identical
    // value in every lane of the half); wave total = lane0 + lane16.
    float t = 0.0f;
    #pragma unroll
    for (int r = 0; r < 8; ++r) {
        const float dp = dpos[r];
        t = fmaf(dp, dp, t);                       // loss_pos term: d_pos^2
        const float h = fmaxf(0.0f, 1.0f - sqrtf(dneg[r]));
        t = fmaf(h, h, t);                         // loss_neg hinge term
    }
    const float wave_total = __shfl(t, 0, 32) + __shfl(t, 16, 32);

    // Deterministic block reduction (ordered, no float atomics).
    __shared__ float smem[8];
    if (lane == 0) smem[wave] = wave_total;
    __syncthreads();
    if (tid == 0) {
        float s = 0.0f;
        #pragma unroll
        for (int w = 0; w < 8; ++w) s += smem[w];
        partials[blockIdx.x] = s;
    }
}

__global__ void ccl_final(const float* __restrict__ partials,
                          float* __restrict__ out, int nblocks)
{
    if (threadIdx.x == 0) {
        float s = 0.0f;
        for (int i = 0; i < nblocks; ++i) s += partials[i];
        // loss = (lp + ln) / (2 * 2B) * 1e6 ; 2*2B = 65536 for B = 16384
        out[0] = s * (1000000.0f / 65536.0f);
    }
}

extern "C" void kernel_launch(void* const* d_in, const int* in_sizes, int n_in,
                              void* d_out, int out_size, void* d_ws, size_t ws_size,
                              hipStream_t stream) {
    const float* fixf = (const float*)d_in[0];
    const float* movf = (const float*)d_in[1];
    const int*   fpts = (const int*)d_in[2];
    const int*   ppts = (const int*)d_in[3];
    const int*   npts = (const int*)d_in[4];
    float* out      = (float*)d_out;
    float* partials = (float*)d_ws;

    const int B      = in_sizes[2] / 3;   // 16384 points
    const int blocks = B / 128;           // 128 points per 256-thread block

    ccl_main<<<blocks, 256, 0, stream>>>(fixf, movf, fpts, ppts, npts, partials);
    ccl_final<<<1, 32, 0, stream>>>(partials, out, blocks);
}